// SparseUnpooler_20074677142317
// MI455X (gfx1250) — compile-verified
//
#include <hip/hip_runtime.h>

typedef __attribute__((ext_vector_type(2))) float v2f;
typedef __attribute__((ext_vector_type(8))) float v8f;

#define N_NODES 10000
#define AS1 8
#define AS2 8
#define NE 320000
#define NB 32
#define SIZE1 (N_NODES * AS1) // 80000
#define SIZE2 (N_NODES * AS2) // 80000

// out[b, s] = bias[s] for all b — re-run every launch (harness poisons d_out).
__global__ void bias_init_kernel(const float* __restrict__ bias,
                                 float* __restrict__ out) {
    int s = blockIdx.x * blockDim.x + threadIdx.x;
    if (s < SIZE2) out[(size_t)blockIdx.y * SIZE2 + s] = bias[s];
}

// One edge per wave32: C(32x8) = Xgather(32x8) * V_e(8x8) via 4x v_wmma_f32_16x16x4_f32,
// then scatter-accumulate with global_atomic_add_f32.
__global__ __launch_bounds__(256) void sparse_unpool_wmma_kernel(
    const float* __restrict__ x,       // (B, SIZE1)
    const float* __restrict__ values,  // (E, 8, 8)
    const int*   __restrict__ indices, // (2, E): [0,:]=rows, [1,:]=cols
    float*       __restrict__ out) {   // (B, SIZE2)
    const int e = blockIdx.x * (blockDim.x >> 5) + (threadIdx.x >> 5);
    if (e >= NE) return;                    // wave-uniform

    const int lane = threadIdx.x & 31;
    const int n    = lane & 15;             // N / M-row index within half-wave
    const int hi   = lane >> 4;             // 0: lanes 0-15, 1: lanes 16-31

    const int row = indices[e];             // output node
    const int col = indices[NE + e];        // input node

    // ---- B operands: V_e padded to 4x16 (cols 8..15 are zero) ----
    // B layout (4x16, K x N): lane L holds N = L&15; vgpr0 = row K=2*hi, vgpr1 = row K=2*hi+1.
    const float* __restrict__ V = values + (size_t)e * (AS1 * AS2);
    v2f b0 = {0.0f, 0.0f};                  // K-chunk 0 (K = 0..3)
    v2f b1 = {0.0f, 0.0f};                  // K-chunk 1 (K = 4..7)
    if (n < AS2) {
        const int k0 = 2 * hi;
        b0.x = V[(k0 + 0) * AS2 + n];
        b0.y = V[(k0 + 1) * AS2 + n];
        b1.x = V[(k0 + 4) * AS2 + n];
        b1.y = V[(k0 + 5) * AS2 + n];
    }

    // ---- A operands: Xg[m][k] = x[batch, col*8 + k] ----
    // A layout (16x4, M x K): lane L holds M = L&15; vgpr0 = K=2*hi, vgpr1 = K=2*hi+1.
    const float* __restrict__ x0 = x + (size_t)n * SIZE1 + (size_t)col * AS1; // batches 0..15
    const float* __restrict__ x1 = x0 + (size_t)16 * SIZE1;                   // batches 16..31
    const int kb0 = 2 * hi;          // K offsets {0,1} or {2,3}
    const int kb1 = 4 + 2 * hi;      // K offsets {4,5} or {6,7}

    v2f a00 = *(const v2f*)(x0 + kb0);   // half 0, K-chunk 0 (8B-aligned)
    v2f a01 = *(const v2f*)(x0 + kb1);   // half 0, K-chunk 1
    v2f a10 = *(const v2f*)(x1 + kb0);   // half 1, K-chunk 0
    v2f a11 = *(const v2f*)(x1 + kb1);   // half 1, K-chunk 1

    v8f c0 = {};   // batches 0..15
    v8f c1 = {};   // batches 16..31
    // (neg_a, A, neg_b, B, c_mod, C, reuse_a, reuse_b)
    c0 = __builtin_amdgcn_wmma_f32_16x16x4_f32(false, a00, false, b0, (short)0, c0, false, false);
    c0 = __builtin_amdgcn_wmma_f32_16x16x4_f32(false, a01, false, b1, (short)0, c0, false, false);
    c1 = __builtin_amdgcn_wmma_f32_16x16x4_f32(false, a10, false, b0, (short)0, c1, false, false);
    c1 = __builtin_amdgcn_wmma_f32_16x16x4_f32(false, a11, false, b1, (short)0, c1, false, false);

    // ---- Scatter-add: C layout: lane L, vgpr r -> C[8*hi + r][L&15] ----
    if (n < AS2) {
        float* obase = out + (size_t)row * AS2 + n;
        const int m0 = hi * 8;
#pragma unroll
        for (int r = 0; r < 8; ++r) {
            atomicAdd(obase + (size_t)(m0 + r)      * SIZE2, c0[r]);
            atomicAdd(obase + (size_t)(16 + m0 + r) * SIZE2, c1[r]);
        }
    }
}

extern "C" void kernel_launch(void* const* d_in, const int* in_sizes, int n_in,
                              void* d_out, int out_size, void* d_ws, size_t ws_size,
                              hipStream_t stream) {
    // setup_inputs() order: x, values, b, indices
    const float* x       = (const float*)d_in[0];
    const float* values  = (const float*)d_in[1];
    const float* bias    = (const float*)d_in[2];
    const int*   indices = (const int*)d_in[3];
    float*       out     = (float*)d_out;

    dim3 gb((SIZE2 + 255) / 256, NB);
    bias_init_kernel<<<gb, 256, 0, stream>>>(bias, out);

    const int wavesPerBlock = 8; // 256 threads = 8 wave32 on CDNA5
    const int blocks = (NE + wavesPerBlock - 1) / wavesPerBlock; // 40000, exact
    sparse_unpool_wmma_kernel<<<blocks, wavesPerBlock * 32, 0, stream>>>(x, values, indices, out);
}